// CausalTransition_962072674910
// MI455X (gfx1250) — compile-verified
//
#include <hip/hip_runtime.h>

typedef float v2f __attribute__((ext_vector_type(2)));
typedef float v8f __attribute__((ext_vector_type(8)));

#define BATCH  2
#define CDIM   64
#define NPIX   256     // H*W
#define NNODE  257     // NPIX + action node
#define NPAD   272     // 17*16 padding for WMMA tiles
#define LDIM   800
#define GWAVES 8       // waves (16x16 tiles) per 256-thread block

// ---------------- workspace layout (floats, 1KB-aligned chunks) -------------
#define OFF_NODES   0L                                    // B*272*64
#define OFF_C0      34816L                                // 800 (pad 1024)
#define OFF_AI      35840L                                // B*272*800
#define OFF_AJ      471040L
#define OFF_XL      906240L
#define OFF_XR      1341440L
#define OFF_ADJ     1776640L                              // B*257*257 (pad)
#define OFF_SCORET  1908992L                              // B*272*272
#define OFF_ALPHAT  2056960L
#define OFF_RELU    2204928L                              // B*272*800
#define OFF_Y       2640128L                              // B*272*64

// ---------------- nodes = [pl ; ba ; zero-pad] ------------------------------
__global__ __launch_bounds__(256) void nodes_kernel(const float* __restrict__ latent,
                                                    const float* __restrict__ ba,
                                                    float* __restrict__ nodes) {
  int idx = blockIdx.x * blockDim.x + threadIdx.x;
  if (idx >= BATCH * NPAD * CDIM) return;
  int c = idx & (CDIM - 1);
  int n = (idx >> 6) % NPAD;
  int b = idx / (NPAD * CDIM);
  float val;
  if (n < NPIX) {
    int k = c >> 1;
    const float coef = -9.210340371976184f / (float)CDIM; // -ln(10000)/C
    float div = expf((float)(2 * k) * coef);
    float ang = (float)n * div;
    float pe = (c & 1) ? cosf(ang) : sinf(ang);
    val = latent[((long)b * CDIM + c) * NPIX + n] + pe;
  } else if (n == NPIX) {
    val = ba[c];
  } else {
    val = 0.0f;
  }
  nodes[idx] = val;
}

// ---------------- c0[h] = ba @ W1a + b1 -------------------------------------
__global__ __launch_bounds__(256) void c0_kernel(const float* __restrict__ ba,
                                                 const float* __restrict__ W1a,
                                                 const float* __restrict__ b1,
                                                 float* __restrict__ c0) {
  int h = blockIdx.x * blockDim.x + threadIdx.x;
  if (h >= LDIM) return;
  float acc = b1[h];
  #pragma unroll 8
  for (int c = 0; c < CDIM; ++c) acc = fmaf(ba[c], W1a[c * LDIM + h], acc);
  c0[h] = acc;
}

// ---------------- generic batched fp32 WMMA GEMM ----------------------------
// C[m,n] = sum_k A[m,k]*B[k,n] (+bias[n]) (+relu). M%16==0, N%16==0, K%4==0.
__global__ __launch_bounds__(256) void gemm_wmma_f32(
    const float* __restrict__ A, const float* __restrict__ B, float* __restrict__ C,
    const float* __restrict__ bias,
    int M, int N, int K, int lda, int ldb, int ldc,
    long sA, long sB, long sC, int act) {
  const int lane = threadIdx.x & 31;
  const int wave = threadIdx.x >> 5;
  const int tilesN = N >> 4;
  const int tilesM = M >> 4;
  const int tile = blockIdx.x * GWAVES + wave;
  if (tile >= tilesM * tilesN) return;            // uniform per wave
  const int tM = tile / tilesN;
  const int tN = tile - tM * tilesN;
  const long b = blockIdx.y;
  A += b * sA; B += b * sB; C += b * sC;

  const int l15 = lane & 15;
  const int hi = lane >> 4;                        // 0 or 1
  const int mRow = tM * 16 + l15;
  const int nCol = tN * 16 + l15;
  const int kOff = hi * 2;

  const float* aPtr = A + (long)mRow * lda + kOff;
  const float* bPtr = B + (long)kOff * ldb + nCol;

  v8f acc = {};
  for (int k = 0; k < K; k += 4) {
    v2f af; af.x = aPtr[0]; af.y = aPtr[1];        // A[16x4] fragment
    v2f bf; bf.x = bPtr[0]; bf.y = bPtr[ldb];      // B[4x16] fragment
    aPtr += 4;
    bPtr += (long)4 * ldb;
    acc = __builtin_amdgcn_wmma_f32_16x16x4_f32(false, af, false, bf,
                                                (short)0, acc, false, false);
  }
  float bv = bias ? bias[nCol] : 0.0f;
  const int mBase = tM * 16 + hi * 8;
  float* cPtr = C + (long)mBase * ldc + nCol;
  #pragma unroll
  for (int r = 0; r < 8; ++r) {
    float v = acc[r] + bv;
    if (act) v = fmaxf(v, 0.0f);
    cPtr[(long)r * ldc] = v;
  }
}

// ---- fused edge scorer + Gumbel hard sample; writes adj (B,257,257) --------
__global__ __launch_bounds__(256) void coeffs_causal_kernel(
    const float* __restrict__ ai, const float* __restrict__ aj,
    const float* __restrict__ c0, const float* __restrict__ w2,
    const float* __restrict__ b2, const float* __restrict__ u,
    float* __restrict__ adj) {
  const int i = blockIdx.x;          // 0..256
  const int b = blockIdx.y;
  float* adjb = adj + (long)b * NNODE * NNODE;
  if (i == NPIX) {                   // border row: all ones
    for (int j = threadIdx.x; j < NNODE; j += 256) adjb[(long)NPIX * NNODE + j] = 1.0f;
    return;
  }
  __shared__ float sAi[LDIM];
  __shared__ float sW2[LDIM];
  const float* aiRow = ai + ((long)b * NPAD + i) * LDIM;
  for (int h = threadIdx.x; h < LDIM; h += 256) { sAi[h] = aiRow[h] + c0[h]; sW2[h] = w2[h]; }
  __syncthreads();

  const int j = threadIdx.x;         // 0..255
  const float* ajRow = aj + ((long)b * NPAD + j) * LDIM;
  float acc = 0.0f;
  #pragma unroll 4
  for (int h = 0; h < LDIM; ++h) {
    float t = sAi[h] + ajRow[h];
    t = (t > 0.0f) ? t : 0.01f * t;                // leaky_relu 0.01
    acc = fmaf(sW2[h], t, acc);
  }
  float v = acc + b2[0];
  float p1 = 1.0f / (1.0f + expf(-v));             // sigmoid
  float l1 = logf(fmaxf(p1, 1e-4f));
  float l0 = logf(fmaxf(1.0f - p1, 1e-4f));
  const float* up = u + (((long)b * NPIX + i) * NPIX + j) * 2;
  float z0 = l0 - logf(-logf(up[0]));
  float z1 = l1 - logf(-logf(up[1]));
  float m = fmaxf(z0, z1);
  float e0 = expf(z0 - m), e1 = expf(z1 - m);
  float y1 = e1 / (e0 + e1);
  float hard = (z1 > z0) ? 1.0f : 0.0f;            // argmax ties -> index 0
  float causal = (hard + y1) - y1;                 // straight-through, fp-exact
  adjb[(long)i * NNODE + j] = causal;
  if (threadIdx.x == 0) adjb[(long)i * NNODE + NPIX] = 1.0f;  // border col
}

// ---- fused GAT score, masked; scoreT[b][t][s] -----------------------------
__global__ __launch_bounds__(256) void score_kernel(
    const float* __restrict__ xl, const float* __restrict__ xr,
    const float* __restrict__ adj, const float* __restrict__ we,
    const float* __restrict__ att, float* __restrict__ scoreT) {
  const int t = blockIdx.x;          // 0..256
  const int b = blockIdx.y;
  __shared__ float sXr[LDIM];
  __shared__ float sWe[LDIM];
  __shared__ float sAtt[LDIM];
  const float* xrRow = xr + ((long)b * NPAD + t) * LDIM;
  for (int h = threadIdx.x; h < LDIM; h += 256) { sXr[h] = xrRow[h]; sWe[h] = we[h]; sAtt[h] = att[h]; }
  __syncthreads();
  const float* adjb = adj + (long)b * NNODE * NNODE;
  float* srow = scoreT + ((long)b * NPAD + t) * NPAD;
  for (int s = threadIdx.x; s < NNODE; s += 256) {
    float a = adjb[(long)s * NNODE + t];
    float sc = -1e30f;
    if (a > 0.0f) {
      const float* xlRow = xl + ((long)b * NPAD + s) * LDIM;
      float acc = 0.0f;
      #pragma unroll 4
      for (int h = 0; h < LDIM; ++h) {
        float f = xlRow[h] + sXr[h] + a * sWe[h];
        f = (f > 0.0f) ? f : 0.2f * f;             // leaky_relu 0.2
        acc = fmaf(sAtt[h], f, acc);
      }
      sc = acc;
    }
    srow[s] = sc;
  }
}

// ---- softmax over s; alphaT[b][t][s], zero-padded to 272 in s -------------
__global__ __launch_bounds__(256) void alpha_kernel(const float* __restrict__ scoreT,
                                                    float* __restrict__ alphaT) {
  const int t = blockIdx.x;
  const int b = blockIdx.y;
  const int tid = threadIdx.x;
  const float* srow = scoreT + ((long)b * NPAD + t) * NPAD;
  float* arow = alphaT + ((long)b * NPAD + t) * NPAD;
  __shared__ float red[256];
  float m = -3.4e38f;
  for (int s = tid; s < NNODE; s += 256) m = fmaxf(m, srow[s]);
  red[tid] = m; __syncthreads();
  for (int o = 128; o > 0; o >>= 1) { if (tid < o) red[tid] = fmaxf(red[tid], red[tid + o]); __syncthreads(); }
  m = red[0];
  __syncthreads();
  float ssum = 0.0f;
  for (int s = tid; s < NNODE; s += 256) ssum += expf(srow[s] - m);
  red[tid] = ssum; __syncthreads();
  for (int o = 128; o > 0; o >>= 1) { if (tid < o) red[tid] += red[tid + o]; __syncthreads(); }
  float inv = 1.0f / red[0];
  for (int s = tid; s < NPAD; s += 256)
    arow[s] = (s < NNODE) ? expf(srow[s] - m) * inv : 0.0f;
}

// ---- (B, t, C) -> (B, C, H, W) --------------------------------------------
__global__ __launch_bounds__(256) void out_kernel(const float* __restrict__ y,
                                                  float* __restrict__ out) {
  int idx = blockIdx.x * blockDim.x + threadIdx.x;
  if (idx >= BATCH * CDIM * NPIX) return;
  int n = idx & (NPIX - 1);
  int c = (idx >> 8) & (CDIM - 1);
  int b = idx >> 14;
  out[idx] = y[((long)b * NPAD + n) * CDIM + c];
}

extern "C" void kernel_launch(void* const* d_in, const int* in_sizes, int n_in,
                              void* d_out, int out_size, void* d_ws, size_t ws_size,
                              hipStream_t stream) {
  (void)in_sizes; (void)n_in; (void)out_size; (void)ws_size;
  const float* latent   = (const float*)d_in[0];
  const float* u        = (const float*)d_in[1];
  // d_in[2] = Wa: dead (multiplied by zeros in reference)
  const float* ba       = (const float*)d_in[3];
  const float* W1i      = (const float*)d_in[4];
  const float* W1j      = (const float*)d_in[5];
  const float* W1a      = (const float*)d_in[6];
  const float* b1       = (const float*)d_in[7];
  const float* w2       = (const float*)d_in[8];
  const float* b2       = (const float*)d_in[9];
  const float* Wl       = (const float*)d_in[10];
  const float* bl       = (const float*)d_in[11];
  const float* Wr       = (const float*)d_in[12];
  const float* br       = (const float*)d_in[13];
  const float* we       = (const float*)d_in[14];
  const float* att      = (const float*)d_in[15];
  const float* gat_bias = (const float*)d_in[16];
  const float* Wo       = (const float*)d_in[17];
  const float* bo       = (const float*)d_in[18];

  float* ws     = (float*)d_ws;
  float* nodes  = ws + OFF_NODES;
  float* c0     = ws + OFF_C0;
  float* ai     = ws + OFF_AI;
  float* aj     = ws + OFF_AJ;
  float* xl     = ws + OFF_XL;
  float* xr     = ws + OFF_XR;
  float* adj    = ws + OFF_ADJ;
  float* scoreT = ws + OFF_SCORET;
  float* alphaT = ws + OFF_ALPHAT;
  float* relu_o = ws + OFF_RELU;
  float* yout   = ws + OFF_Y;

  nodes_kernel<<<(BATCH * NPAD * CDIM + 255) / 256, 256, 0, stream>>>(latent, ba, nodes);
  c0_kernel<<<(LDIM + 255) / 256, 256, 0, stream>>>(ba, W1a, b1, c0);

  // G1: nodes (B,272,64) @ {W1i,W1j,Wl,Wr} (64,800)  -> ai/aj/xl/xr (B,272,800)
  {
    int tiles = (NPAD / 16) * (LDIM / 16);               // 850
    dim3 g((tiles + GWAVES - 1) / GWAVES, BATCH);
    long sA = (long)NPAD * CDIM, sC = (long)NPAD * LDIM;
    gemm_wmma_f32<<<g, 256, 0, stream>>>(nodes, W1i, ai, nullptr, NPAD, LDIM, CDIM, CDIM, LDIM, LDIM, sA, 0, sC, 0);
    gemm_wmma_f32<<<g, 256, 0, stream>>>(nodes, W1j, aj, nullptr, NPAD, LDIM, CDIM, CDIM, LDIM, LDIM, sA, 0, sC, 0);
    gemm_wmma_f32<<<g, 256, 0, stream>>>(nodes, Wl,  xl, bl,      NPAD, LDIM, CDIM, CDIM, LDIM, LDIM, sA, 0, sC, 0);
    gemm_wmma_f32<<<g, 256, 0, stream>>>(nodes, Wr,  xr, br,      NPAD, LDIM, CDIM, CDIM, LDIM, LDIM, sA, 0, sC, 0);
  }

  coeffs_causal_kernel<<<dim3(NNODE, BATCH), 256, 0, stream>>>(ai, aj, c0, w2, b2, u, adj);
  score_kernel<<<dim3(NNODE, BATCH), 256, 0, stream>>>(xl, xr, adj, we, att, scoreT);
  alpha_kernel<<<dim3(NNODE, BATCH), 256, 0, stream>>>(scoreT, alphaT);

  // G2: relu(alphaT (B,272,272) @ xl (B,272,800) + gat_bias)
  {
    int tiles = (NPAD / 16) * (LDIM / 16);
    dim3 g((tiles + GWAVES - 1) / GWAVES, BATCH);
    gemm_wmma_f32<<<g, 256, 0, stream>>>(alphaT, xl, relu_o, gat_bias,
                                         NPAD, LDIM, NPAD, NPAD, LDIM, LDIM,
                                         (long)NPAD * NPAD, (long)NPAD * LDIM, (long)NPAD * LDIM, 1);
  }
  // G3: relu_o (B,272,800) @ Wo (800,64) + bo -> yout (B,272,64)
  {
    int tiles = (NPAD / 16) * (CDIM / 16);               // 68
    dim3 g((tiles + GWAVES - 1) / GWAVES, BATCH);
    gemm_wmma_f32<<<g, 256, 0, stream>>>(relu_o, Wo, yout, bo,
                                         NPAD, CDIM, LDIM, LDIM, CDIM, CDIM,
                                         (long)NPAD * LDIM, 0, (long)NPAD * CDIM, 0);
  }

  out_kernel<<<(BATCH * CDIM * NPIX + 255) / 256, 256, 0, stream>>>(yout, (float*)d_out);
}